// TransformerEncoderLayer_68264210202662
// MI455X (gfx1250) — compile-verified
//
#include <hip/hip_runtime.h>

// ---------------------------------------------------------------------------
// TransformerEncoderLayer with relative-position (RBF x SphHarm) attention,
// MI455X / gfx1250 (wave32, WMMA).
//
// Key algebraic optimization: rp_emb (B,L,L,512) is never materialized.
//   rp_scores[b,h,l,m] = feat[b,l,m,:576] . qw[b,l,h,:576]
//   qw[b,l,h,k]        = sum_d q[b,l,h,d] * rp_w[h*64+d, k]
// (the rp_b contribution is constant in m and cancels in softmax)
// This turns a 121-GFLOP GEMM + 419MB intermediate into ~2.3 GFLOP, all
// operands L2-resident (192MB L2). The 576-length pairwise contraction and
// both FFN GEMMs run on V_WMMA_F32_16X16X4_F32. FFN GEMM waves compute
// 16x64 tiles (4 chained accumulators) so each A fragment feeds 4 WMMAs,
// with global_prefetch_b8 pulling the next k-chunk of the B streams.
// ---------------------------------------------------------------------------

#define BATCH 2
#define SEQ   320
#define DIM   512
#define HEADS 8
#define DHEAD 64
#define NRAD  64
#define SPH   9
#define RP    576                      // 64 * 9
#define DELTA (10.0f / 63.0f)
#define GAMMA ((63.0f / 10.0f) * (63.0f / 10.0f))
#define SCALE 0.125f                   // 1/sqrt(64)

typedef __attribute__((ext_vector_type(2))) float v2f;
typedef __attribute__((ext_vector_type(8))) float v8f;

__device__ __forceinline__ float block_reduce_sum(float v, float* red) {
  const int tid = threadIdx.x;
  red[tid] = v;
  __syncthreads();
  for (int s = 128; s > 0; s >>= 1) {
    if (tid < s) red[tid] += red[tid + s];
    __syncthreads();
  }
  float r = red[0];
  __syncthreads();
  return r;
}

// One block per (b, l) query row. 256 threads = 8 waves.
__global__ __launch_bounds__(256)
void attn_rp_kernel(const float* __restrict__ src,
                    const float* __restrict__ rel_diss,
                    const float* __restrict__ rel_dirs,
                    const float* __restrict__ rp_w,
                    const float* __restrict__ ln1_g,
                    const float* __restrict__ ln1_b,
                    float* __restrict__ xbuf) {
  __shared__ float q_l[DIM];                 //  2.0 KB  query row
  __shared__ float qw[HEADS][RP];            // 18.4 KB  q . rp_w (per head)
  __shared__ float feat[8][16][36];          // 18.4 KB  per-wave feat chunk
  __shared__ float sphb[8][16][SPH];         //  4.6 KB  spherical harmonics
  __shared__ float disb[8][16];              //  0.5 KB
  __shared__ float scores[HEADS][SEQ];       // 10.2 KB
  __shared__ float outrow[DIM];              //  2.0 KB
  __shared__ float red[256];                 //  1.0 KB   (total ~57 KB LDS)

  const int b    = blockIdx.x / SEQ;
  const int l    = blockIdx.x % SEQ;
  const int tid  = threadIdx.x;
  const int wave = tid >> 5;
  const int lane = tid & 31;
  const int half = lane >> 4;   // which K-pair this lane holds in a WMMA frag
  const int low  = lane & 15;   // M row (A / D) or N col (B / D)

  // ---- load query row ----
  for (int i = tid; i < DIM; i += 256) q_l[i] = src[(size_t)(b * SEQ + l) * DIM + i];
  __syncthreads();

  // ---- qw[h][k] = sum_d q_l[h*64+d] * rp_w[(h*64+d)*576 + k]  (rp_w in L2)
  for (int idx = tid; idx < HEADS * RP; idx += 256) {
    const int h = idx / RP;
    const int k = idx - h * RP;
    const float* w = rp_w + (size_t)(h * DHEAD) * RP + k;
    const float* q = q_l + h * DHEAD;
    float acc = 0.f;
#pragma unroll 8
    for (int d = 0; d < DHEAD; ++d) acc = fmaf(q[d], w[(size_t)d * RP], acc);
    qw[h][k] = acc;
  }
  __syncthreads();

  // ---- scores: each wave owns 16-row m-tiles; D = feat(16x576) . qw^T(576x16)
  for (int tile = wave; tile < SEQ / 16; tile += 8) {
    const int m0 = tile * 16;

    if (lane < 16) {   // geometry for the 16 m's of this tile
      const int m = m0 + lane;
      const size_t pair = (size_t)(b * SEQ + l) * SEQ + m;
      const float dd = rel_diss[pair];
      const float* dp = rel_dirs + pair * 3;
      const float dx = dp[0], dy = dp[1], dz = dp[2];
      disb[wave][lane] = dd;
      sphb[wave][lane][0] = 0.28209479177387814f;
      sphb[wave][lane][1] = 0.4886025119029199f * dy;
      sphb[wave][lane][2] = 0.4886025119029199f * dz;
      sphb[wave][lane][3] = 0.4886025119029199f * dx;
      sphb[wave][lane][4] = 1.0925484305920792f * dx * dy;
      sphb[wave][lane][5] = 1.0925484305920792f * dy * dz;
      sphb[wave][lane][6] = 0.31539156525252005f * (3.f * dz * dz - 1.f);
      sphb[wave][lane][7] = 1.0925484305920792f * dx * dz;
      sphb[wave][lane][8] = 0.5462742152960396f * (dx * dx - dy * dy);
    }
    __builtin_amdgcn_wave_barrier();

    v8f acc = {0.f, 0.f, 0.f, 0.f, 0.f, 0.f, 0.f, 0.f};

    // K=576 processed in 16 chunks of 4 radials (36 k-values, 9 WMMA steps)
    for (int r0 = 0; r0 < NRAD; r0 += 4) {
      // fill feat[wave][m][rr*9+s]; 16m x 4r pairs -> 2 per lane; 1 exp each
#pragma unroll
      for (int p = 0; p < 2; ++p) {
        const int pi = lane + 32 * p;
        const int mm = pi & 15;
        const int rr = pi >> 4;            // 0..3
        const float dd = disb[wave][mm];
        const float t  = dd - (float)(r0 + rr) * DELTA;
        const float e  = __expf(-GAMMA * t * t);
#pragma unroll
        for (int s = 0; s < SPH; ++s)
          feat[wave][mm][rr * SPH + s] = e * sphb[wave][mm][s];
      }
      __builtin_amdgcn_wave_barrier();

      const int kbase = r0 * SPH;
#pragma unroll
      for (int ks = 0; ks < 9; ++ks) {
        const int kl = ks * 4 + 2 * half;                 // lane's K pair
        v2f a = *(const v2f*)&feat[wave][low][kl];        // A: 16 x 4 f32
        v2f bb = {0.f, 0.f};
        if (low < HEADS) bb = *(const v2f*)&qw[low][kbase + kl];  // B: 4 x 16
        acc = __builtin_amdgcn_wmma_f32_16x16x4_f32(
            false, a, false, bb, (short)0, acc, false, false);
      }
      __builtin_amdgcn_wave_barrier();
    }

    // add base q.q score (head-block-diagonal -> VALU) and spill tile
    if (low < HEADS) {
      const int h = low;
      const float* qh = q_l + h * DHEAD;
#pragma unroll
      for (int v = 0; v < 8; ++v) {
        const int m = m0 + v + 8 * half;                  // D-matrix row layout
        const float* qm = src + (size_t)(b * SEQ + m) * DIM + h * DHEAD;
        float base = 0.f;
#pragma unroll 8
        for (int d = 0; d < DHEAD; ++d) base = fmaf(qh[d], qm[d], base);
        scores[h][m] = acc[v] + SCALE * base;
      }
    }
  }
  __syncthreads();

  // ---- softmax over m, one wave per head ----
  {
    const int h = wave;
    float mx = -3.4e38f;
    for (int m = lane; m < SEQ; m += 32) mx = fmaxf(mx, scores[h][m]);
    for (int off = 16; off > 0; off >>= 1) mx = fmaxf(mx, __shfl_xor(mx, off, 32));
    float sum = 0.f;
    for (int m = lane; m < SEQ; m += 32) {
      const float e = __expf(scores[h][m] - mx);
      scores[h][m] = e;
      sum += e;
    }
    for (int off = 16; off > 0; off >>= 1) sum += __shfl_xor(sum, off, 32);
    const float inv = 1.f / sum;
    for (int m = lane; m < SEQ; m += 32) scores[h][m] *= inv;
  }
  __syncthreads();

  // ---- out[:,h,d] = sum_m attn[h][m] * src[b,m,h*64+d]  (coalesced, L2) ----
  for (int idx = tid; idx < DIM; idx += 256) {
    const int h = idx >> 6;
    const float* col = src + (size_t)b * SEQ * DIM + idx;
    float acc = 0.f;
    for (int m = 0; m < SEQ; ++m) acc = fmaf(scores[h][m], col[(size_t)m * DIM], acc);
    outrow[idx] = acc;
  }
  __syncthreads();

  // ---- residual + LayerNorm1 -> x ----
  float part = 0.f, part2 = 0.f;
  for (int i = tid; i < DIM; i += 256) {
    const float y = q_l[i] + outrow[i];
    outrow[i] = y;
    part += y;
    part2 += y * y;
  }
  const float mean = block_reduce_sum(part, red) * (1.f / DIM);
  const float m2   = block_reduce_sum(part2, red) * (1.f / DIM);
  const float rstd = rsqrtf(m2 - mean * mean + 1e-5f);
  for (int i = tid; i < DIM; i += 256)
    xbuf[(size_t)(b * SEQ + l) * DIM + i] = (outrow[i] - mean) * rstd * ln1_g[i] + ln1_b[i];
}

// h = leaky_relu(x @ w1.T + b1). Each wave computes a 16x64 tile: one A frag
// feeds 4 chained f32 WMMAs; B streams prefetched one 64-k chunk ahead.
__global__ __launch_bounds__(256)
void ffn_gemm1_kernel(const float* __restrict__ x,    // (640,512)
                      const float* __restrict__ w1,   // (2048,512)
                      const float* __restrict__ b1,
                      float* __restrict__ hbuf) {     // (640,2048)
  const int N = 4 * DIM, K = DIM;
  const int wave = threadIdx.x >> 5, lane = threadIdx.x & 31;
  const int half = lane >> 4, low = lane & 15;
  const int tile = blockIdx.x * 8 + wave;              // 16x64 tiles
  const int tn = tile % (N / 64), tm = tile / (N / 64);
  const int m0 = tm * 16, n0 = tn * 64;
  const float* arow = x + (size_t)(m0 + low) * K;
  const float* brow0 = w1 + (size_t)(n0 + low) * K;    // B[k][n] = w1[n][k]
  const float* brow1 = brow0 + (size_t)16 * K;
  const float* brow2 = brow0 + (size_t)32 * K;
  const float* brow3 = brow0 + (size_t)48 * K;
  v8f acc0 = {0.f, 0.f, 0.f, 0.f, 0.f, 0.f, 0.f, 0.f};
  v8f acc1 = acc0, acc2 = acc0, acc3 = acc0;
  for (int k0 = 0; k0 < K; k0 += 64) {
    if (k0 + 64 < K) {                                  // next-chunk prefetch
      __builtin_prefetch(arow + k0 + 64 + 2 * half, 0, 1);
      __builtin_prefetch(brow0 + k0 + 64 + 2 * half, 0, 1);
      __builtin_prefetch(brow1 + k0 + 64 + 2 * half, 0, 1);
      __builtin_prefetch(brow2 + k0 + 64 + 2 * half, 0, 1);
      __builtin_prefetch(brow3 + k0 + 64 + 2 * half, 0, 1);
    }
#pragma unroll 4
    for (int k = k0; k < k0 + 64; k += 4) {
      const int ko = k + 2 * half;
      v2f a  = *(const v2f*)(arow + ko);
      v2f b0 = *(const v2f*)(brow0 + ko);
      v2f b1f = *(const v2f*)(brow1 + ko);
      v2f b2f = *(const v2f*)(brow2 + ko);
      v2f b3f = *(const v2f*)(brow3 + ko);
      acc0 = __builtin_amdgcn_wmma_f32_16x16x4_f32(false, a, false, b0, (short)0, acc0, false, false);
      acc1 = __builtin_amdgcn_wmma_f32_16x16x4_f32(false, a, false, b1f, (short)0, acc1, false, false);
      acc2 = __builtin_amdgcn_wmma_f32_16x16x4_f32(false, a, false, b2f, (short)0, acc2, false, false);
      acc3 = __builtin_amdgcn_wmma_f32_16x16x4_f32(false, a, false, b3f, (short)0, acc3, false, false);
    }
  }
#pragma unroll
  for (int j = 0; j < 4; ++j) {
    const v8f acc = (j == 0) ? acc0 : (j == 1) ? acc1 : (j == 2) ? acc2 : acc3;
    const int n = n0 + j * 16 + low;
    const float bias = b1[n];
#pragma unroll
    for (int v = 0; v < 8; ++v) {
      const int m = m0 + v + 8 * half;
      const float t = acc[v] + bias;
      hbuf[(size_t)m * N + n] = (t > 0.f) ? t : 0.01f * t;
    }
  }
}

// y = h @ w2.T + b2 + x (residual). 16x64 tiles, f32 WMMA, K=2048.
__global__ __launch_bounds__(256)
void ffn_gemm2_kernel(const float* __restrict__ hbuf,  // (640,2048)
                      const float* __restrict__ w2,    // (512,2048)
                      const float* __restrict__ b2,
                      const float* __restrict__ x,     // (640,512)
                      float* __restrict__ ybuf) {      // (640,512)
  const int N = DIM, K = 4 * DIM;
  const int wave = threadIdx.x >> 5, lane = threadIdx.x & 31;
  const int half = lane >> 4, low = lane & 15;
  const int tile = blockIdx.x * 8 + wave;
  const int tn = tile % (N / 64), tm = tile / (N / 64);
  const int m0 = tm * 16, n0 = tn * 64;
  const float* arow = hbuf + (size_t)(m0 + low) * K;
  const float* brow0 = w2 + (size_t)(n0 + low) * K;
  const float* brow1 = brow0 + (size_t)16 * K;
  const float* brow2 = brow0 + (size_t)32 * K;
  const float* brow3 = brow0 + (size_t)48 * K;
  v8f acc0 = {0.f, 0.f, 0.f, 0.f, 0.f, 0.f, 0.f, 0.f};
  v8f acc1 = acc0, acc2 = acc0, acc3 = acc0;
  for (int k0 = 0; k0 < K; k0 += 64) {
    if (k0 + 64 < K) {
      __builtin_prefetch(arow + k0 + 64 + 2 * half, 0, 1);
      __builtin_prefetch(brow0 + k0 + 64 + 2 * half, 0, 1);
      __builtin_prefetch(brow1 + k0 + 64 + 2 * half, 0, 1);
      __builtin_prefetch(brow2 + k0 + 64 + 2 * half, 0, 1);
      __builtin_prefetch(brow3 + k0 + 64 + 2 * half, 0, 1);
    }
#pragma unroll 4
    for (int k = k0; k < k0 + 64; k += 4) {
      const int ko = k + 2 * half;
      v2f a  = *(const v2f*)(arow + ko);
      v2f b0 = *(const v2f*)(brow0 + ko);
      v2f b1f = *(const v2f*)(brow1 + ko);
      v2f b2f = *(const v2f*)(brow2 + ko);
      v2f b3f = *(const v2f*)(brow3 + ko);
      acc0 = __builtin_amdgcn_wmma_f32_16x16x4_f32(false, a, false, b0, (short)0, acc0, false, false);
      acc1 = __builtin_amdgcn_wmma_f32_16x16x4_f32(false, a, false, b1f, (short)0, acc1, false, false);
      acc2 = __builtin_amdgcn_wmma_f32_16x16x4_f32(false, a, false, b2f, (short)0, acc2, false, false);
      acc3 = __builtin_amdgcn_wmma_f32_16x16x4_f32(false, a, false, b3f, (short)0, acc3, false, false);
    }
  }
#pragma unroll
  for (int j = 0; j < 4; ++j) {
    const v8f acc = (j == 0) ? acc0 : (j == 1) ? acc1 : (j == 2) ? acc2 : acc3;
    const int n = n0 + j * 16 + low;
    const float bias = b2[n];
#pragma unroll
    for (int v = 0; v < 8; ++v) {
      const int m = m0 + v + 8 * half;
      ybuf[(size_t)m * N + n] = acc[v] + bias + x[(size_t)m * N + n];
    }
  }
}

__global__ __launch_bounds__(256)
void ln2_kernel(const float* __restrict__ y,
                const float* __restrict__ g,
                const float* __restrict__ bta,
                float* __restrict__ out) {
  __shared__ float red[256];
  const int row = blockIdx.x, tid = threadIdx.x;
  const float* yr = y + (size_t)row * DIM;
  float part = 0.f, part2 = 0.f;
  for (int i = tid; i < DIM; i += 256) {
    const float v = yr[i];
    part += v;
    part2 += v * v;
  }
  const float mean = block_reduce_sum(part, red) * (1.f / DIM);
  const float m2   = block_reduce_sum(part2, red) * (1.f / DIM);
  const float rstd = rsqrtf(m2 - mean * mean + 1e-5f);
  for (int i = tid; i < DIM; i += 256)
    out[(size_t)row * DIM + i] = (yr[i] - mean) * rstd * g[i] + bta[i];
}

extern "C" void kernel_launch(void* const* d_in, const int* in_sizes, int n_in,
                              void* d_out, int out_size, void* d_ws, size_t ws_size,
                              hipStream_t stream) {
  (void)in_sizes; (void)n_in; (void)out_size; (void)ws_size;
  const float* src      = (const float*)d_in[0];
  const float* rel_diss = (const float*)d_in[1];
  const float* rel_dirs = (const float*)d_in[2];
  const float* rp_w     = (const float*)d_in[3];
  // d_in[4] = rp_b: contributes a constant (per b,h,l) to the scores -> cancels in softmax
  const float* w1       = (const float*)d_in[5];
  const float* b1       = (const float*)d_in[6];
  const float* w2       = (const float*)d_in[7];
  const float* b2       = (const float*)d_in[8];
  const float* ln1_g    = (const float*)d_in[9];
  const float* ln1_b    = (const float*)d_in[10];
  const float* ln2_g    = (const float*)d_in[11];
  const float* ln2_b    = (const float*)d_in[12];

  float* xbuf = (float*)d_ws;                         // 640*512
  float* hbuf = xbuf + (size_t)BATCH * SEQ * DIM;     // 640*2048
  float* ybuf = hbuf + (size_t)BATCH * SEQ * 4 * DIM; // 640*512  (~7.9 MB total)

  attn_rp_kernel<<<BATCH * SEQ, 256, 0, stream>>>(src, rel_diss, rel_dirs, rp_w,
                                                  ln1_g, ln1_b, xbuf);
  // 16x64 tiles: (640/16)*(2048/64) = 40*32 = 1280 wave-tiles / 8 = 160 blocks
  ffn_gemm1_kernel<<<(BATCH * SEQ / 16) * (4 * DIM / 64) / 8, 256, 0, stream>>>(
      xbuf, w1, b1, hbuf);
  // (640/16)*(512/64) = 40*8 = 320 wave-tiles / 8 = 40 blocks
  ffn_gemm2_kernel<<<(BATCH * SEQ / 16) * (DIM / 64) / 8, 256, 0, stream>>>(
      hbuf, w2, b2, xbuf, ybuf);
  ln2_kernel<<<BATCH * SEQ, 256, 0, stream>>>(ybuf, ln2_g, ln2_b, (float*)d_out);
}